// CenterAlignment_83502754169266
// MI455X (gfx1250) — compile-verified
//
#include <hip/hip_runtime.h>
#include <hip/hip_bf16.h>

// ---------------------------------------------------------------------------
// CenterAlignment on MI455X (gfx1250), compile-only tuned.
// Phase 1: bandwidth-bound segment-sum (512 MB of x reads -> ~22us floor):
//   - column-sliced LDS accumulation with ds_add_f32
//   - TDM (tensor_load_to_lds) double-buffered label staging + s_wait_tensorcnt
//   - low-volume global_atomic_add_f32 flush
// Phase 2: per-class momentum update + renorm (wave reductions)
// Phase 3: Gram-diagonal ||diff||^2 via v_wmma_f32_16x16x4_f32 (pure f32)
// ---------------------------------------------------------------------------

#define B_ROWS   524288
#define DCOLS    256
#define NCLASS   1000
#define MOM      0.9f

typedef unsigned int       u32;
typedef unsigned long long u64;
typedef u32   v4u __attribute__((ext_vector_type(4)));
typedef int   v4i __attribute__((ext_vector_type(4)));
typedef int   v8i __attribute__((ext_vector_type(8)));
typedef float v2f __attribute__((ext_vector_type(2)));
typedef float v8f __attribute__((ext_vector_type(8)));

// ---------------- K0: zero the workspace accumulators ----------------------
__global__ void zero_ws_kernel(float* __restrict__ p, int n) {
  int i = blockIdx.x * blockDim.x + threadIdx.x;
  if (i < n) p[i] = 0.0f;
}

// ---------------- TDM helper: 1-D copy of nelem i32 words global -> LDS ----
// D# per CDNA5 ISA ch.8: group0 {count=1, lds_addr, global_addr57, type=2},
// group1 {data_size=4B, tensor_dim0=nelem, tensor_dim1=1, tile_dim0=nelem,
//         tensor_dim0_stride=nelem}, groups 2/3 zero (<=2D tensor).
// This toolchain's builtin takes 6 args (extra zero-filled v8i before cpol).
__device__ __forceinline__ void tdm_load_i32(const int* gsrc, u32 lds_byte, int nelem) {
  u64 ga = (u64)(uintptr_t)gsrc;
  v4u g0;
  g0[0] = 1u;                                    // count=1, user mode
  g0[1] = lds_byte;                              // lds_addr (bytes)
  g0[2] = (u32)ga;                               // global_addr[31:0]
  g0[3] = (u32)((ga >> 32) & 0x01FFFFFFull)      // global_addr[56:32]
          | (2u << 30);                          // type=2 ("image")
  v8i g1;
  g1[0] = (2 << 16);                             // data_size=2 (4 bytes)
  g1[1] = (nelem & 0xFFFF) << 16;                // tensor_dim0[15:0]   @ bits63:48
  g1[2] = ((nelem >> 16) & 0xFFFF) | (1 << 16);  // tensor_dim0[31:16], tensor_dim1=1
  g1[3] = (nelem & 0xFFFF) << 16;                // tile_dim0           @ bits127:112
  g1[4] = 0;                                     // tile_dim1=0, tile_dim2=0
  g1[5] = nelem;                                 // tensor_dim0_stride[31:0]
  g1[6] = 0;
  g1[7] = 0;
  v4i z4 = {0, 0, 0, 0};
  v8i z8 = {0, 0, 0, 0, 0, 0, 0, 0};
  __builtin_amdgcn_tensor_load_to_lds(g0, g1, z4, z4, z8, 0);
}

// ---------------- K1: segment-sum of x into [C,D] + counts -----------------
#define NSLICE        8
#define WGS_PER_SLICE 128
#define ROWS_PER_WG   (B_ROWS / WGS_PER_SLICE)   // 4096
#define LCHUNK        2048
#define NCHUNK        (ROWS_PER_WG / LCHUNK)     // 2
#define ACC_FLOATS    (NCLASS * 32)              // 32000
#define CNT_OFF       ACC_FLOATS
#define LBL_OFF       (ACC_FLOATS + NCLASS)      // 33000 floats
#define K1_SMEM_BYTES (LBL_OFF * 4 + 2 * LCHUNK * 4)  // 148384 B

__global__ __launch_bounds__(256, 1) void segsum_kernel(
    const float* __restrict__ x, const int* __restrict__ l,
    float* __restrict__ sums, float* __restrict__ counts) {
  extern __shared__ float smem[];
  float* acc = smem;                       // [1000][32]
  float* cnt = smem + CNT_OFF;             // [1000]
  int*   lbl = (int*)(smem + LBL_OFF);     // [2][LCHUNK]

  const int tid   = threadIdx.x;
  const int lane  = tid & 31;
  const int wave  = tid >> 5;
  const int slice = blockIdx.x & (NSLICE - 1);
  const int wgs   = blockIdx.x >> 3;
  const long long row_base = (long long)wgs * ROWS_PER_WG;

  // Kick off first label DMA (wave 0 only), overlap with LDS zero-init.
  if (tid < 32) {
    tdm_load_i32(l + row_base, (u32)(uintptr_t)lbl, LCHUNK);
  }
  for (int i = tid; i < LBL_OFF; i += 256) smem[i] = 0.0f;

  const float* xp = x + row_base * DCOLS + slice * 32 + lane;

  for (int c = 0; c < NCHUNK; ++c) {
    if (tid < 32) {
      if (c + 1 < NCHUNK) {
        tdm_load_i32(l + row_base + (long long)(c + 1) * LCHUNK,
                     (u32)(uintptr_t)(lbl + ((c + 1) & 1) * LCHUNK), LCHUNK);
        __builtin_amdgcn_s_wait_tensorcnt(1);   // chunk c done (in-order TDM)
      } else {
        __builtin_amdgcn_s_wait_tensorcnt(0);
      }
    }
    __syncthreads();

    const int*   lb = lbl + (c & 1) * LCHUNK;
    const float* xc = xp + (long long)c * LCHUNK * DCOLS;
    #pragma unroll 4
    for (int i = wave; i < LCHUNK; i += 8) {
      int   lab = lb[i];                          // LDS broadcast load
      float v   = xc[(long long)i * DCOLS];       // coalesced 128B/wave
      atomicAdd(&acc[lab * 32 + lane], v);        // ds_add_f32
      if (slice == 0 && lane == 0) atomicAdd(&cnt[lab], 1.0f);
    }
    __syncthreads();
  }

  // Flush LDS partials to global with f32 atomics (low volume).
  for (int i = tid; i < ACC_FLOATS; i += 256) {
    int cls = i >> 5;
    int col = slice * 32 + (i & 31);
    unsafeAtomicAdd(&sums[cls * DCOLS + col], acc[i]);
  }
  if (slice == 0) {
    for (int i = tid; i < NCLASS; i += 256) unsafeAtomicAdd(&counts[i], cnt[i]);
  }
}

// ---------------- K2a: momentum update + L2 renorm -> new_img --------------
__global__ __launch_bounds__(256) void update_centers_kernel(
    const float* __restrict__ sums, const float* __restrict__ counts,
    const float* __restrict__ center_img, float* __restrict__ new_img) {
  const int wave = threadIdx.x >> 5, lane = threadIdx.x & 31;
  const int cls = blockIdx.x * 8 + wave;          // grid=125 -> cls<1000
  const float cn = counts[cls];
  const bool present = cn > 0.0f;
  const float inv = 1.0f / fmaxf(cn, 1.0f);
  float u[8], ci[8], ss = 0.0f;
  #pragma unroll
  for (int j = 0; j < 8; ++j) {
    int col = lane + 32 * j;
    ci[j] = center_img[cls * DCOLS + col];
    float mean = sums[cls * DCOLS + col] * inv;
    u[j] = MOM * ci[j] + (1.0f - MOM) * mean;
    ss += u[j] * u[j];
  }
  #pragma unroll
  for (int off = 16; off; off >>= 1) ss += __shfl_xor(ss, off, 32);
  const float rn = 1.0f / sqrtf(ss);
  #pragma unroll
  for (int j = 0; j < 8; ++j) {
    int col = lane + 32 * j;
    new_img[cls * DCOLS + col] = present ? u[j] * rn : ci[j];
  }
}

// ---------------- K2b: masked sum of ||diff||^2 via WMMA Gram diagonal -----
// One wave handles 16 classes: D = A * A^T with A = diff tile (16 x 4 per
// step, K accumulated over 256). The 16x4 f32 A striping and 4x16 f32 B
// striping coincide element-wise when B = A^T, so the same registers are
// passed for both operands. Diagonal of the 16x16 f32 C/D tile: m<8 ->
// VGPR m lane m; m>=8 -> VGPR m-8 lane m+16.
__global__ __launch_bounds__(32) void loss_wmma_kernel(
    const float* __restrict__ new_img, const float* __restrict__ skt,
    const float* __restrict__ counts, float* __restrict__ scal) {
  __shared__ float sdiff[16 * DCOLS];
  const int lane = threadIdx.x;
  const int base = blockIdx.x * 16;

  for (int idx = lane; idx < 16 * DCOLS; idx += 32) {
    int m = idx >> 8, col = idx & (DCOLS - 1);
    int cls = base + m;
    float d = 0.0f;
    if (cls < NCLASS) d = new_img[cls * DCOLS + col] - skt[cls * DCOLS + col];
    sdiff[idx] = d;
  }
  __syncthreads();

  const int lr = lane & 15, half = lane >> 4;
  v8f c = {0.f, 0.f, 0.f, 0.f, 0.f, 0.f, 0.f, 0.f};
  for (int k0 = 0; k0 < DCOLS; k0 += 4) {
    v2f a;                                        // A 16x4 f32 striping:
    a.x = sdiff[lr * DCOLS + k0 + 2 * half];      // VGPR0: K=0 | K=2
    a.y = sdiff[lr * DCOLS + k0 + 1 + 2 * half];  // VGPR1: K=1 | K=3
    c = __builtin_amdgcn_wmma_f32_16x16x4_f32(
        false, a, false, a, (short)0, c, false, false);
  }

  float sq = 0.0f;                                // extract diag(D)
  #pragma unroll
  for (int r = 0; r < 8; ++r) {
    float d0 = __shfl(c[r], r, 32);               // (m=r,   n=r)
    float d1 = __shfl(c[r], r + 24, 32);          // (m=r+8, n=r+8)
    if (lane == r)     sq = d0;
    if (lane == r + 8) sq = d1;
  }

  const int cls = base + lane;
  const bool valid = (lane < 16) && (cls < NCLASS);
  const bool present = valid && (counts[valid ? cls : 0] > 0.0f);
  float contrib = present ? sq : 0.0f;
  #pragma unroll
  for (int off = 16; off; off >>= 1) contrib += __shfl_xor(contrib, off, 32);
  unsigned long long bal = __ballot(present ? 1 : 0);
  if (lane == 0) {
    unsafeAtomicAdd(&scal[0], contrib);
    unsafeAtomicAdd(&scal[1], (float)__popcll(bal));
  }
}

// ---------------- K3: finalize scalar loss ---------------------------------
__global__ void finalize_kernel(const float* __restrict__ scal,
                                float* __restrict__ out) {
  if (threadIdx.x == 0 && blockIdx.x == 0)
    out[0] = scal[0] / fmaxf(scal[1], 1.0f);
}

// ---------------------------------------------------------------------------
extern "C" void kernel_launch(void* const* d_in, const int* in_sizes, int n_in,
                              void* d_out, int out_size, void* d_ws, size_t ws_size,
                              hipStream_t stream) {
  (void)in_sizes; (void)n_in; (void)out_size; (void)ws_size;
  const float* x    = (const float*)d_in[0];
  const int*   l    = (const int*)d_in[1];     // jax default x64 off -> int32
  const float* cimg = (const float*)d_in[2];
  const float* cskt = (const float*)d_in[3];
  float* out = (float*)d_out;

  float* ws      = (float*)d_ws;
  float* sums    = ws;                       // [1000*256]
  float* counts  = ws + NCLASS * DCOLS;      // [1000]
  float* scal    = counts + NCLASS;          // loss_sum, n_present
  float* new_img = scal + 8;                 // [1000*256]
  const int zero_n = NCLASS * DCOLS + NCLASS + 2;

  zero_ws_kernel<<<(zero_n + 255) / 256, 256, 0, stream>>>(ws, zero_n);

  segsum_kernel<<<NSLICE * WGS_PER_SLICE, 256, K1_SMEM_BYTES, stream>>>(
      x, l, sums, counts);

  update_centers_kernel<<<NCLASS / 8, 256, 0, stream>>>(
      sums, counts, cimg, new_img);

  loss_wmma_kernel<<<(NCLASS + 15) / 16, 32, 0, stream>>>(
      new_img, cskt, counts, scal);

  finalize_kernel<<<1, 32, 0, stream>>>(scal, out);
}